// SelectiveSSMLayer_19782619365543
// MI455X (gfx1250) — compile-verified
//
#include <hip/hip_runtime.h>
#include <hip/hip_bf16.h>
#include <stdint.h>

typedef __attribute__((ext_vector_type(2))) float v2f;
typedef __attribute__((ext_vector_type(4))) float f4;
typedef __attribute__((ext_vector_type(8))) float v8f;

// ---------------------------------------------------------------------------
// WMMA f32 GEMM:  C[M,N] = A[M,K] (row major) * W[N,K]^T (row major)
// Block: 256 threads = 8 waves (2 Mwaves x 4 Nwaves). Block tile 64x128.
// Each wave: 2x2 accumulators of 16x16, K-step 16 -> 16 WMMAs per iter.
// LDS rows padded to 20 floats: float4-aligned, conflict-free frag reads.
// Tile staging uses GLOBAL_LOAD_ASYNC_TO_LDS_B128 (ASYNCcnt path).
// ---------------------------------------------------------------------------
__launch_bounds__(256)
__global__ void wmma_gemm_tn(const float* __restrict__ A,
                             const float* __restrict__ W,
                             float* __restrict__ C,
                             int M, int N, int K)
{
    __shared__ alignas(16) float lA[64 * 20];
    __shared__ alignas(16) float lB[128 * 20];

    const int t    = threadIdx.x;
    const int w    = t >> 5;
    const int lane = t & 31;
    const int wm   = w >> 2;        // 0..1
    const int wn   = w & 3;         // 0..3
    const int lr   = lane & 15;     // row within 16-tile
    const int lh   = lane >> 4;     // lane half (K pair select)

    const int m0 = blockIdx.y * 64;
    const int n0 = blockIdx.x * 128;

    // global staging roles
    const int am = t >> 2;          // 0..63   A row in tile
    const int ak = (t & 3) * 4;     // 0,4,8,12
    const int bn = t >> 1;          // 0..127  W row in tile
    const int bk = (t & 1) * 8;     // 0 or 8
    const int gn = n0 + bn;

    // LDS byte addresses for this lane's async-staging slots
    const uint32_t ldsA = (uint32_t)(uintptr_t)&lA[am * 20 + ak];
    const uint32_t ldsB = (uint32_t)(uintptr_t)&lB[bn * 20 + bk];

    // OOB W rows: zero their LDS slots once; those lanes never async-load,
    // so the zeros persist across all K iterations.
    if (gn >= N) {
        f4 zz = {};
        *(f4*)(&lB[bn * 20 + bk])     = zz;
        *(f4*)(&lB[bn * 20 + bk + 4]) = zz;
    }

    v8f acc[2][2] = {};

    for (int k0 = 0; k0 < K; k0 += 16) {
        // async stage A tile [64 x 16]: 256 lanes x 16B, memory -> LDS direct
        const float* ga = A + (size_t)(m0 + am) * K + k0 + ak;
        asm volatile("global_load_async_to_lds_b128 %0, %1, off"
                     :: "v"(ldsA), "v"(ga) : "memory");
        // async stage W tile [128 x 16]: 2 x 16B per lane
        // (INST_OFFSET applies to BOTH the LDS and global address)
        if (gn < N) {
            const float* wp = W + (size_t)gn * K + k0 + bk;
            asm volatile("global_load_async_to_lds_b128 %0, %1, off"
                         :: "v"(ldsB), "v"(wp) : "memory");
            asm volatile("global_load_async_to_lds_b128 %0, %1, off offset:16"
                         :: "v"(ldsB), "v"(wp) : "memory");
        }
        asm volatile("s_wait_asynccnt 0x0" ::: "memory");
        __syncthreads();

#pragma unroll
        for (int ks = 0; ks < 16; ks += 4) {
            v2f afrag[2], bfrag[2];
#pragma unroll
            for (int mt = 0; mt < 2; ++mt)
                afrag[mt] = *(const v2f*)(&lA[(wm * 32 + mt * 16 + lr) * 20 + ks + 2 * lh]);
#pragma unroll
            for (int nt = 0; nt < 2; ++nt)
                bfrag[nt] = *(const v2f*)(&lB[(wn * 32 + nt * 16 + lr) * 20 + ks + 2 * lh]);
#pragma unroll
            for (int mt = 0; mt < 2; ++mt)
#pragma unroll
                for (int nt = 0; nt < 2; ++nt)
                    acc[mt][nt] = __builtin_amdgcn_wmma_f32_16x16x4_f32(
                        false, afrag[mt], false, bfrag[nt],
                        (short)0, acc[mt][nt], false, false);
        }
        __syncthreads();
    }

    // write-back: acc vgpr r, lane l -> M = r + 8*lh, N = lr
#pragma unroll
    for (int mt = 0; mt < 2; ++mt) {
#pragma unroll
        for (int nt = 0; nt < 2; ++nt) {
            const int n = n0 + wn * 32 + nt * 16 + lr;
            if (n < N) {
#pragma unroll
                for (int r = 0; r < 8; ++r) {
                    const int m = m0 + wm * 32 + mt * 16 + r + 8 * lh;
                    C[(size_t)m * N + n] = acc[mt][nt][r];
                }
            }
        }
    }
}

// ---------------------------------------------------------------------------
// Selective scan. One block per (batch, head): 128 blocks, 256 threads.
// State h[32][64] in registers: thread t owns p = t>>3, n = 8*(t&7)..+7.
// proj row layout: [z:0 | x:2048 | Braw:4096 | Craw:4160 | dd_dt:4224 |
//                   dd_A:4288 | trap:4352 | angles:4416] (row len 4432)
// ---------------------------------------------------------------------------
__device__ __forceinline__ float softplusf(float x) {
    return (x > 20.f) ? x : log1pf(expf(x));
}

__launch_bounds__(256)
__global__ void ssm_scan(const float* __restrict__ proj,
                         const float* __restrict__ dt_bias,
                         const float* __restrict__ B_bias,
                         const float* __restrict__ C_bias,
                         const float* __restrict__ Bn_w,
                         const float* __restrict__ Cn_w,
                         const float* __restrict__ Dv,
                         float* __restrict__ yg)
{
    const int blk = blockIdx.x;
    const int b   = blk >> 6;
    const int hd  = blk & 63;
    const int t   = threadIdx.x;

    __shared__ float sB[64], sC[64], sGB[64], sCM[64];
    __shared__ float sx[32], sz[32], sCos[16], sSin[16];
    __shared__ float sScal[4];                  // a, g*dt, rsB, rsC
    __shared__ float sWB[64], sWC[64], sBB[64], sCB[64];

    if (t < 64) {
        sWB[t] = Bn_w[t];
        sWC[t] = Cn_w[t];
        sBB[t] = B_bias[hd * 64 + t];
        sCB[t] = C_bias[hd * 64 + t];
    }
    const float dtb = dt_bias[hd];
    const float Dh  = Dv[hd];

    float hreg[8];
#pragma unroll
    for (int i = 0; i < 8; ++i) hreg[i] = 0.f;

    const int p  = t >> 3;
    const int ng = (t & 7) << 3;
    __syncthreads();

    for (int s = 0; s < 1024; ++s) {
        const float* row  = proj + (size_t)(b * 1024 + s) * 4432;
        const float* rown = row + 4432;

        // ---- phase 1: stage this step's inputs ----
        if (t < 64) {
            sB[t] = row[4096 + t];
            if (t < 11) __builtin_prefetch(rown + 4096 + t * 32, 0, 1);
        } else if (t < 128) {
            sC[t - 64] = row[4160 + (t - 64)];
        } else if (t < 160) {
            sx[t - 128] = row[2048 + hd * 32 + (t - 128)];
        } else if (t < 192) {
            sz[t - 160] = row[hd * 32 + (t - 160)];
        } else if (t < 208) {
            const int j = t - 192;
            const float ang = row[4416 + j];
            sCos[j] = cosf(ang);
            sSin[j] = sinf(ang);
        } else if (t == 208) {
            const float ddt = row[4224 + hd];
            const float dA  = row[4288 + hd];
            const float tr  = row[4352 + hd];
            const float dt  = softplusf(ddt + dtb);
            const float Am  = fminf(-softplusf(dA), -1e-4f);
            const float sg  = 1.f / (1.f + expf(-tr));
            const float g   = 1.f - 0.5f * sg;
            sScal[0] = g * expf(Am * dt) + 0.5f * sg;   // a_t
            sScal[1] = g * dt;                          // g*dt
        } else if (t >= 224 && t < 228) {
            __builtin_prefetch(rown + hd * 32 + (t - 224) * 16, 0, 1);
            __builtin_prefetch(rown + 2048 + hd * 32 + (t - 224) * 16, 0, 1);
        }
        __syncthreads();

        // ---- phase 2: RMS reductions (wave0 -> B, wave1 -> C) ----
        if (t < 32) {
            float v0 = sB[t], v1 = sB[t + 32];
            float ss = v0 * v0 + v1 * v1;
            ss += __shfl_xor(ss, 1);
            ss += __shfl_xor(ss, 2);
            ss += __shfl_xor(ss, 4);
            ss += __shfl_xor(ss, 8);
            ss += __shfl_xor(ss, 16);
            if (t == 0) sScal[2] = rsqrtf(ss * (1.f / 64.f) + 1e-5f);
        } else if (t < 64) {
            const int l = t - 32;
            float v0 = sC[l], v1 = sC[l + 32];
            float ss = v0 * v0 + v1 * v1;
            ss += __shfl_xor(ss, 1);
            ss += __shfl_xor(ss, 2);
            ss += __shfl_xor(ss, 4);
            ss += __shfl_xor(ss, 8);
            ss += __shfl_xor(ss, 16);
            if (l == 0) sScal[3] = rsqrtf(ss * (1.f / 64.f) + 1e-5f);
        }
        __syncthreads();

        // ---- phase 3: norm + bias + RoPE -> g*dt*Bm and Cm ----
        const float a_t = sScal[0];
        const float gdt = sScal[1];
        if (t < 64) {
            const float rs = sScal[2];
            const int n = t;
            float bm;
            if (n < 16) {
                const float v1 = sB[n] * rs * sWB[n] + sBB[n];
                const float v2 = sB[n + 16] * rs * sWB[n + 16] + sBB[n + 16];
                bm = v1 * sCos[n] - v2 * sSin[n];
            } else if (n < 32) {
                const int j = n - 16;
                const float v1 = sB[j] * rs * sWB[j] + sBB[j];
                const float v2 = sB[n] * rs * sWB[n] + sBB[n];
                bm = v1 * sSin[j] + v2 * sCos[j];
            } else {
                bm = sB[n] * rs * sWB[n] + sBB[n];
            }
            sGB[n] = gdt * bm;
        } else if (t < 128) {
            const float rs = sScal[3];
            const int n = t - 64;
            float cm;
            if (n < 16) {
                const float v1 = sC[n] * rs * sWC[n] + sCB[n];
                const float v2 = sC[n + 16] * rs * sWC[n + 16] + sCB[n + 16];
                cm = v1 * sCos[n] - v2 * sSin[n];
            } else if (n < 32) {
                const int j = n - 16;
                const float v1 = sC[j] * rs * sWC[j] + sCB[j];
                const float v2 = sC[n] * rs * sWC[n] + sCB[n];
                cm = v1 * sSin[j] + v2 * sCos[j];
            } else {
                cm = sC[n] * rs * sWC[n] + sCB[n];
            }
            sCM[n] = cm;
        }
        __syncthreads();

        // ---- phase 4: state update + output ----
        const float xp = sx[p];
        float accy = 0.f;
#pragma unroll
        for (int i = 0; i < 8; ++i) {
            const int n = ng + i;
            hreg[i] = a_t * hreg[i] + xp * sGB[n];
            accy += hreg[i] * sCM[n];
        }
        accy += __shfl_xor(accy, 1);
        accy += __shfl_xor(accy, 2);
        accy += __shfl_xor(accy, 4);
        if ((t & 7) == 0) {
            const float y   = accy + Dh * xp;
            const float z   = sz[p];
            const float sig = 1.f / (1.f + expf(-z));
            yg[(size_t)(b * 1024 + s) * 2048 + hd * 32 + p] = y * (z * sig);
        }
        __syncthreads();
    }
}

// ---------------------------------------------------------------------------
// inputs: u, W_in, dt_bias, B_bias, C_bias, Bn_w, Cn_w, D, W_out
// ---------------------------------------------------------------------------
extern "C" void kernel_launch(void* const* d_in, const int* in_sizes, int n_in,
                              void* d_out, int out_size, void* d_ws, size_t ws_size,
                              hipStream_t stream)
{
    const float* u       = (const float*)d_in[0];
    const float* W_in    = (const float*)d_in[1];
    const float* dt_bias = (const float*)d_in[2];
    const float* B_bias  = (const float*)d_in[3];
    const float* C_bias  = (const float*)d_in[4];
    const float* Bn_w    = (const float*)d_in[5];
    const float* Cn_w    = (const float*)d_in[6];
    const float* Dv      = (const float*)d_in[7];
    const float* W_out   = (const float*)d_in[8];
    float*       out     = (float*)d_out;

    float* proj = (float*)d_ws;                        // [2048, 4432]
    float* yg   = proj + (size_t)2048 * 4432;          // [2048, 2048]

    dim3 blk(256);
    // proj = u @ W_in^T   (M=2048, N=4432, K=1024)
    wmma_gemm_tn<<<dim3(35, 32), blk, 0, stream>>>(u, W_in, proj, 2048, 4432, 1024);
    // sequential selective scan + gating
    ssm_scan<<<dim3(128), blk, 0, stream>>>(proj, dt_bias, B_bias, C_bias,
                                            Bn_w, Cn_w, Dv, yg);
    // out = yg @ W_out^T  (M=2048, N=1024, K=2048)
    wmma_gemm_tn<<<dim3(8, 32), blk, 0, stream>>>(yg, W_out, out, 2048, 1024, 2048);
}